// SOBEL_18743237279946
// MI455X (gfx1250) — compile-verified
//
#include <hip/hip_runtime.h>
#include <cstdint>
#include <cstddef>

// ---------------------------------------------------------------------------
// Sobel L1 edge loss, MI455X (gfx1250).
// loss = |Gx(pred)-Gx(gt)| + |Gy(pred)-Gy(gt)|, 48 x 512 x 512 fp32 images.
// Pure HBM-bandwidth kernel (~151 MB -> ~6.5us floor @ 23.3 TB/s).
// CDNA5 path used: GLOBAL_LOAD_ASYNC_TO_LDS_B128 staging (ASYNCcnt),
// LDS-halo stencil, b128 non-temporal stores. WMMA intentionally not used:
// 0.25 GFLOP total, matrix layout shuffles would only add traffic.
// ---------------------------------------------------------------------------

typedef __attribute__((ext_vector_type(4))) float v4f;

#define NIMG        48
#define HH          512
#define WW          512
#define TILE_W      128          // output tile width  (32 x float4)
#define TILE_H      32           // output tile height
#define LROW        136          // floats per LDS row: 128 + 4-col halo each side
#define LROWS       34           // 32 rows + 1-row halo each side
#define SEG_PER_ROW 34           // float4 segments per LDS row
#define NSEG        (LROWS * SEG_PER_ROW)   // 1156 segments per buffer
#define NTHREADS    256          // 8 wave32s

// Async DMA of one 16B segment global -> LDS. Per-lane, tracked by ASYNCcnt.
// Inline asm (not the clang builtin) so it works on both ROCm-7.2 and
// amdgpu-toolchain regardless of builtin arity.
__device__ __forceinline__ void async_copy_b128(const float* gsrc, const float* ldst) {
  // Generic pointer to a __shared__ object: addr[31:0] is the LDS byte offset
  // (flat-aperture mapping, ISA 10.2).
  unsigned lds_off = (unsigned)(size_t)ldst;
  unsigned long long gaddr = (unsigned long long)(size_t)gsrc;
  asm volatile("global_load_async_to_lds_b128 %0, %1, off"
               :: "v"(lds_off), "v"(gaddr)
               : "memory");
}

__global__ __launch_bounds__(NTHREADS) void sobel_l1_kernel(
    const float* __restrict__ pred, const float* __restrict__ gt,
    float* __restrict__ out) {
  __shared__ v4f ldsPv[NSEG];
  __shared__ v4f ldsGv[NSEG];
  float* ldsP = (float*)ldsPv;
  float* ldsG = (float*)ldsGv;

  const int tid = threadIdx.x;
  const int img = blockIdx.z;
  const int ty0 = blockIdx.y * TILE_H;   // first output row of tile
  const int tx0 = blockIdx.x * TILE_W;   // first output col of tile
  const size_t imgBase = (size_t)img * (HH * WW);

  // ---- Phase 1: stage pred/gt tiles (with halo) into LDS via async DMA ----
  // Segment s -> LDS row lr (global row ty0-1+lr), float4 col lc4
  // (global col tx0-4+4*lc4). Tiles are 128-aligned, so every halo segment is
  // either fully inside the image or fully outside (-> zero pad).
  for (int s = tid; s < NSEG; s += NTHREADS) {
    const int lr  = s / SEG_PER_ROW;
    const int lc4 = s - lr * SEG_PER_ROW;
    const int gy  = ty0 - 1 + lr;
    const int gx  = tx0 - 4 + lc4 * 4;
    float* lp = &ldsP[lr * LROW + lc4 * 4];
    float* lg = &ldsG[lr * LROW + lc4 * 4];
    if ((unsigned)gy < (unsigned)HH && (unsigned)gx < (unsigned)WW) {
      const size_t off = imgBase + (size_t)gy * WW + (size_t)gx;
      async_copy_b128(pred + off, lp);
      async_copy_b128(gt   + off, lg);
    } else {
      v4f z = 0.0f;                       // zero padding
      *(v4f*)lp = z;
      *(v4f*)lg = z;
    }
  }
  asm volatile("s_wait_asynccnt 0" ::: "memory");  // our DMAs landed in LDS
  __syncthreads();                                  // everyone's DMAs landed

  // ---- Phase 2: stencil from LDS ----
  // Thread -> float4 column tx (0..31) and row group tz (0..7); computes a
  // 4-wide x 4-tall output micro-tile with a rolling 3x6 register window.
  const int tx  = tid & 31;
  const int tz  = tid >> 5;
  const int y0  = tz * 4;                 // first output row (tile-local)
  const int col = tx * 4 + 3;             // LDS col of (x-1) for first pixel

  // d = pred - gt rows; loss = |Gx(d)| + |Gy(d)| by linearity of the stencil.
  float top[6], mid[6], bot[6];
#pragma unroll
  for (int j = 0; j < 6; ++j) {
    top[j] = ldsP[(y0 + 0) * LROW + col + j] - ldsG[(y0 + 0) * LROW + col + j];
    mid[j] = ldsP[(y0 + 1) * LROW + col + j] - ldsG[(y0 + 1) * LROW + col + j];
    bot[j] = ldsP[(y0 + 2) * LROW + col + j] - ldsG[(y0 + 2) * LROW + col + j];
  }

  v4f* outv = (v4f*)(out + imgBase);
  const size_t ocol = (size_t)(tx0 / 4 + tx);

#pragma unroll
  for (int r = 0; r < 4; ++r) {
    // cross-correlation with KX rows {1,0,-1;2,0,-2;1,0,-1}:
    //   Gx = (1,2,1)-weighted column at x-1  minus  same at x+1
    //   Gy = (1,2,1)-weighted row    at y-1  minus  same at y+1
    float cs[6];
#pragma unroll
    for (int j = 0; j < 6; ++j) cs[j] = top[j] + 2.0f * mid[j] + bot[j];

    v4f o;
#pragma unroll
    for (int k = 0; k < 4; ++k) {
      const float gxv = cs[k] - cs[k + 2];
      const float gyv = (top[k] + 2.0f * top[k + 1] + top[k + 2])
                      - (bot[k] + 2.0f * bot[k + 1] + bot[k + 2]);
      o[k] = __builtin_fabsf(gxv) + __builtin_fabsf(gyv);
    }

    const size_t gyo = (size_t)(ty0 + y0 + r);
    __builtin_nontemporal_store(o, &outv[gyo * (WW / 4) + ocol]);  // write-once

    if (r < 3) {
      const int nr = (y0 + 3 + r) * LROW + col;
#pragma unroll
      for (int j = 0; j < 6; ++j) {
        top[j] = mid[j];
        mid[j] = bot[j];
        bot[j] = ldsP[nr + j] - ldsG[nr + j];
      }
    }
  }
}

extern "C" void kernel_launch(void* const* d_in, const int* in_sizes, int n_in,
                              void* d_out, int out_size, void* d_ws, size_t ws_size,
                              hipStream_t stream) {
  const float* pred = (const float*)d_in[0];
  const float* gt   = (const float*)d_in[1];
  float* out        = (float*)d_out;
  (void)in_sizes; (void)n_in; (void)out_size; (void)d_ws; (void)ws_size;

  dim3 grid(WW / TILE_W, HH / TILE_H, NIMG);   // 4 x 16 x 48 = 3072 blocks
  dim3 block(NTHREADS, 1, 1);
  sobel_l1_kernel<<<grid, block, 0, stream>>>(pred, gt, out);
}